// GCN_72533407695321
// MI455X (gfx1250) — compile-verified
//
#include <hip/hip_runtime.h>
#include <hip/hip_bf16.h>

#define N_NODES   200000
#define N_EDGES   6400000
#define F_IN      128
#define N_CLASSES 10
#define N_GRAPHS  64

typedef float v2f __attribute__((ext_vector_type(2)));
typedef float v8f __attribute__((ext_vector_type(8)));

// ---------------- degree / dinv ----------------

__global__ void gcn_deg_init(float* __restrict__ deg, int n) {
    int i = blockIdx.x * blockDim.x + threadIdx.x;
    if (i < n) deg[i] = 1.0f;   // self-loop weight
}

__global__ void gcn_deg_scatter(const int* __restrict__ dst,
                                const float* __restrict__ ew,
                                float* __restrict__ deg, int e) {
    int i = blockIdx.x * blockDim.x + threadIdx.x;
    if (i < e) atomicAdd(&deg[dst[i]], ew[i]);
}

__global__ void gcn_dinv(float* __restrict__ deg, int n) {
    int i = blockIdx.x * blockDim.x + threadIdx.x;
    if (i < n) deg[i] = __frsqrt_rn(deg[i]);   // deg >= 1 always
}

// ---------------- fp32 WMMA GEMM:  C[nrows x NO] = A[nrows x K] @ B[K x NO] ----------------
// One wave computes one 16x16 output tile with V_WMMA_F32_16X16X4_F32, K-loop step 4.
// A frag (16x4): lane = M row (lane&15); VGPR0/VGPR1 hold K = kh, kh+1 where
// kh = 0 for lanes 0-15, kh = 2 for lanes 16-31 (ISA 7.12.2 32-bit A layout).
// B frag (4x16): lane = N col, same K split. C/D: VGPR j -> M = j + 8*(lane>>4), N = lane&15.
template <int K, int NO>
__global__ void gcn_wmma_gemm(const float* __restrict__ A,
                              const float* __restrict__ B,
                              float* __restrict__ C, int nrows) {
    const int colTiles = NO / 16;
    int wave = (blockIdx.x * blockDim.x + threadIdx.x) >> 5;
    int lane = threadIdx.x & 31;
    int totalTiles = (nrows >> 4) * colTiles;
    if (wave >= totalTiles) return;
    int tr = wave / colTiles;
    int tc = wave - tr * colTiles;

    int m    = lane & 15;
    int kh   = (lane >> 4) << 1;          // 0 or 2
    int half = lane >> 4;

    const float* Arow = A + (size_t)(tr * 16 + m) * K;
    const float* Bcol = B + tc * 16 + m;

    v8f acc = {};
#pragma unroll
    for (int kk = 0; kk < K; kk += 4) {
        v2f a, b;
        a.x = Arow[kk + kh];
        a.y = Arow[kk + kh + 1];
        b.x = Bcol[(size_t)(kk + kh) * NO];
        b.y = Bcol[(size_t)(kk + kh + 1) * NO];
        acc = __builtin_amdgcn_wmma_f32_16x16x4_f32(
            /*neg_a=*/false, a, /*neg_b=*/false, b,
            /*c_mod=*/(short)0, acc, /*reuse_a=*/false, /*reuse_b=*/false);
    }

    float* Crow = C + (size_t)(tr * 16) * NO + tc * 16 + m;
#pragma unroll
    for (int j = 0; j < 8; ++j)
        Crow[(size_t)(j + half * 8) * NO] = acc[j];
}

// ---------------- aggregation: out[i] = dinv[i]^2 * h[i]  (self loop term) ----------------
template <int D>
__global__ void gcn_self_init(const float* __restrict__ h,
                              const float* __restrict__ dinv,
                              float* __restrict__ out, int n) {
    const int CH = D / 4;
    long long t = (long long)blockIdx.x * blockDim.x + threadIdx.x;
    int i = (int)(t / CH);
    int c = (int)(t % CH);
    if (i >= n) return;
    float s = dinv[i] * dinv[i];
    float4 v = *(const float4*)(h + (size_t)i * D + c * 4);
    float4 o; o.x = s * v.x; o.y = s * v.y; o.z = s * v.z; o.w = s * v.w;
    *(float4*)(out + (size_t)i * D + c * 4) = o;
}

// ---------------- aggregation: edge scatter with atomics ----------------
template <int D>
__global__ void gcn_edge_scatter(const int* __restrict__ src,
                                 const int* __restrict__ dst,
                                 const float* __restrict__ ew,
                                 const float* __restrict__ dinv,
                                 const float* __restrict__ h,
                                 float* __restrict__ out, int e) {
    const int CH = D / 4;
    long long t = (long long)blockIdx.x * blockDim.x + threadIdx.x;
    int i = (int)(t / CH);
    int c = (int)(t % CH);
    if (i >= e) return;
    int s = src[i], d = dst[i];
    float coef = dinv[s] * ew[i] * dinv[d];
    float4 v = *(const float4*)(h + (size_t)s * D + c * 4);
    float* o = out + (size_t)d * D + c * 4;
    atomicAdd(o + 0, coef * v.x);
    atomicAdd(o + 1, coef * v.y);
    atomicAdd(o + 2, coef * v.z);
    atomicAdd(o + 3, coef * v.w);
}

// ---------------- bias + relu ----------------
template <int D>
__global__ void gcn_bias_relu(float* __restrict__ out,
                              const float* __restrict__ b, int n) {
    long long t = (long long)blockIdx.x * blockDim.x + threadIdx.x;
    if (t >= (long long)n * D) return;
    int f = (int)(t % D);
    float v = out[t] + b[f];
    out[t] = v > 0.0f ? v : 0.0f;
}

// ---------------- pooling ----------------
__global__ void gcn_pool_init(float* __restrict__ sums, float* __restrict__ cnt) {
    int i = blockIdx.x * blockDim.x + threadIdx.x;
    if (i < N_GRAPHS * 64) sums[i] = 0.0f;
    if (i < N_GRAPHS) cnt[i] = 0.0f;
}

__global__ void gcn_pool(const float* __restrict__ h, const int* __restrict__ batch,
                         float* __restrict__ sums, float* __restrict__ cnt, int n) {
    long long t = (long long)blockIdx.x * blockDim.x + threadIdx.x;
    int i = (int)(t / 16);
    int c = (int)(t % 16);
    if (i >= n) return;
    int g = batch[i];
    float4 v = *(const float4*)(h + (size_t)i * 64 + c * 4);
    float* s = sums + g * 64 + c * 4;
    atomicAdd(s + 0, v.x);
    atomicAdd(s + 1, v.y);
    atomicAdd(s + 2, v.z);
    atomicAdd(s + 3, v.w);
    if (c == 0) atomicAdd(&cnt[g], 1.0f);
}

__global__ void gcn_final(const float* __restrict__ sums, const float* __restrict__ cnt,
                          const float* __restrict__ Wf, const float* __restrict__ bf,
                          float* __restrict__ out) {
    int g = blockIdx.x;
    int c = threadIdx.x;
    if (c >= N_CLASSES) return;
    float inv = 1.0f / fmaxf(cnt[g], 1.0f);
    float acc = bf[c];
#pragma unroll 8
    for (int f = 0; f < 64; ++f)
        acc += sums[g * 64 + f] * inv * Wf[f * N_CLASSES + c];
    out[g * N_CLASSES + c] = acc;
}

// ---------------- launcher ----------------
static inline int cdiv(long long a, long long b) { return (int)((a + b - 1) / b); }

extern "C" void kernel_launch(void* const* d_in, const int* in_sizes, int n_in,
                              void* d_out, int out_size, void* d_ws, size_t ws_size,
                              hipStream_t stream) {
    (void)in_sizes; (void)n_in; (void)out_size; (void)ws_size;

    const float* x   = (const float*)d_in[0];
    const int*   ei  = (const int*)d_in[1];            // [2, E] int32
    const float* ew  = (const float*)d_in[2];
    const int*   bat = (const int*)d_in[3];
    const float* W1  = (const float*)d_in[4];
    const float* b1  = (const float*)d_in[5];
    const float* W2  = (const float*)d_in[6];
    const float* b2  = (const float*)d_in[7];
    const float* W3  = (const float*)d_in[8];
    const float* b3  = (const float*)d_in[9];
    const float* Wf  = (const float*)d_in[10];
    const float* bf  = (const float*)d_in[11];
    float* out = (float*)d_out;

    const int*   src = ei;
    const int*   dst = ei + N_EDGES;

    // workspace layout (floats)
    float* ws   = (float*)d_ws;
    float* dinv = ws;                                   // N
    float* bufA = dinv + N_NODES;                       // N*64
    float* bufB = bufA + (size_t)N_NODES * 64;          // N*64
    float* sums = bufB + (size_t)N_NODES * 64;          // 64*64
    float* cnt  = sums + N_GRAPHS * 64;                 // 64

    const int T = 256;

    // ---- degree -> dinv ----
    gcn_deg_init<<<cdiv(N_NODES, T), T, 0, stream>>>(dinv, N_NODES);
    gcn_deg_scatter<<<cdiv(N_EDGES, T), T, 0, stream>>>(dst, ew, dinv, N_EDGES);
    gcn_dinv<<<cdiv(N_NODES, T), T, 0, stream>>>(dinv, N_NODES);

    // ---- layer 1: 128 -> 16 ----
    {
        const int tiles = (N_NODES / 16) * (16 / 16);
        gcn_wmma_gemm<128, 16><<<cdiv((long long)tiles * 32, T), T, 0, stream>>>(x, W1, bufA, N_NODES);
        gcn_self_init<16><<<cdiv((long long)N_NODES * 4, T), T, 0, stream>>>(bufA, dinv, bufB, N_NODES);
        gcn_edge_scatter<16><<<cdiv((long long)N_EDGES * 4, T), T, 0, stream>>>(src, dst, ew, dinv, bufA, bufB, N_EDGES);
        gcn_bias_relu<16><<<cdiv((long long)N_NODES * 16, T), T, 0, stream>>>(bufB, b1, N_NODES);
    }

    // ---- layer 2: 16 -> 32 ----
    {
        const int tiles = (N_NODES / 16) * (32 / 16);
        gcn_wmma_gemm<16, 32><<<cdiv((long long)tiles * 32, T), T, 0, stream>>>(bufB, W2, bufA, N_NODES);
        gcn_self_init<32><<<cdiv((long long)N_NODES * 8, T), T, 0, stream>>>(bufA, dinv, bufB, N_NODES);
        gcn_edge_scatter<32><<<cdiv((long long)N_EDGES * 8, T), T, 0, stream>>>(src, dst, ew, dinv, bufA, bufB, N_EDGES);
        gcn_bias_relu<32><<<cdiv((long long)N_NODES * 32, T), T, 0, stream>>>(bufB, b2, N_NODES);
    }

    // ---- layer 3: 32 -> 64 ----
    {
        const int tiles = (N_NODES / 16) * (64 / 16);
        gcn_wmma_gemm<32, 64><<<cdiv((long long)tiles * 32, T), T, 0, stream>>>(bufB, W3, bufA, N_NODES);
        gcn_self_init<64><<<cdiv((long long)N_NODES * 16, T), T, 0, stream>>>(bufA, dinv, bufB, N_NODES);
        gcn_edge_scatter<64><<<cdiv((long long)N_EDGES * 16, T), T, 0, stream>>>(src, dst, ew, dinv, bufA, bufB, N_EDGES);
        gcn_bias_relu<64><<<cdiv((long long)N_NODES * 64, T), T, 0, stream>>>(bufB, b3, N_NODES);
    }

    // ---- mean pool + classifier ----
    gcn_pool_init<<<cdiv(N_GRAPHS * 64, 64), 64, 0, stream>>>(sums, cnt);
    gcn_pool<<<cdiv((long long)N_NODES * 16, T), T, 0, stream>>>(bufB, bat, sums, cnt, N_NODES);
    gcn_final<<<N_GRAPHS, 16, 0, stream>>>(sums, cnt, Wf, bf, out);
}